// NoiseLearnModule_56693568307247
// MI455X (gfx1250) — compile-verified
//
#include <hip/hip_runtime.h>
#include <stdint.h>

// NoiseLearnModule: out = x + valid * eps * sigmoid(params[f, bin(x)]) * 0.1
// Shapes (from reference): x/eps [B=65536, F=256] fp32, bins [F, 17], params [F*16].
// Pure streaming op: 192 MB HBM traffic -> ~8.2 us floor at 23.3 TB/s.
// CDNA5 paths used: wave32, async global->LDS DMA staging (ASYNCcnt), LDS table.

constexpr int   F  = 256;
constexpr int   NB = 16;
constexpr int   NE = NB + 1;          // 17 edges per feature
constexpr float NOISE_SCALE = 0.1f;
constexpr int   ROWS = 4;             // rows per block-iteration (MLP unroll)

__global__ __launch_bounds__(F) void noise_learn_kernel(
    const float* __restrict__ x,
    const float* __restrict__ bins,
    const float* __restrict__ eps,
    const float* __restrict__ params,
    float* __restrict__ out,
    int B)
{
    __shared__ float s_bins[F * NE];    // staged copy of bin edges, [f][j] row-major
    __shared__ float s_scale[F * NE];   // fused 0.1*sigmoid table, stride 17 (bank-conflict-free)

    const int tid = threadIdx.x;        // one thread <-> one feature column

    // ---- Stage bin edges into LDS via gfx1250 async global->LDS DMA ----
    {
        const uint32_t lds0 = (uint32_t)(uintptr_t)&s_bins[0];  // low 32 bits = LDS offset
#pragma unroll
        for (int k = 0; k < NE; ++k) {
            const int i = k * F + tid;                          // coalesced across lanes
            const uint32_t ldsAddr = lds0 + 4u * (uint32_t)i;
            const float* gaddr = bins + i;
            asm volatile("global_load_async_to_lds_b32 %0, %1, off"
                         :: "v"(ldsAddr), "v"(gaddr) : "memory");
        }
#if __has_builtin(__builtin_amdgcn_s_wait_asynccnt)
        __builtin_amdgcn_s_wait_asynccnt(0);
#else
        asm volatile("s_wait_asynccnt 0" ::: "memory");
#endif
    }
    __syncthreads();

    // ---- Per-feature precompute: edges -> registers, scale table -> LDS ----
    float e[NE];
#pragma unroll
    for (int j = 0; j < NE; ++j) e[j] = s_bins[tid * NE + j];

#pragma unroll
    for (int j = 0; j < NB; ++j) {
        const float p = params[tid * NB + j];                   // tiny, L2-hot, one-time
        s_scale[tid * NE + j] = NOISE_SCALE / (1.0f + expf(-p));
    }
    // scale table is produced and consumed by the same thread; LDS ops are
    // in-order within a wave, so no extra barrier is required.

    // ---- Streaming main loop: grid-stride over groups of ROWS rows ----
    const int rowStride = (int)gridDim.x * ROWS;
    for (int r0 = (int)blockIdx.x * ROWS; r0 + ROWS <= B; r0 += rowStride) {
        const size_t base = (size_t)r0 * F + (size_t)tid;
        float xv[ROWS], ev[ROWS];
#pragma unroll
        for (int rr = 0; rr < ROWS; ++rr) {
            xv[rr] = __builtin_nontemporal_load(x   + base + (size_t)rr * F);
            ev[rr] = __builtin_nontemporal_load(eps + base + (size_t)rr * F);
        }
#pragma unroll
        for (int rr = 0; rr < ROWS; ++rr) {
            // searchsorted(side='right'): count of edges <= x, minus 1
            int cnt = 0;
#pragma unroll
            for (int j = 0; j < NE; ++j) cnt += (xv[rr] >= e[j]) ? 1 : 0;
            const int idx = cnt - 1;                         // [-1, 16]
            int jc = idx < 0 ? 0 : (idx > NB - 1 ? NB - 1 : idx);
            const float s = s_scale[tid * NE + jc];          // dynamic LDS lookup
            const float noise = ((unsigned)idx < (unsigned)NB) ? ev[rr] * s : 0.0f;
            __builtin_nontemporal_store(xv[rr] + noise, out + base + (size_t)rr * F);
        }
    }

    // ---- Tail rows (B not a multiple of ROWS) ----
    for (int r = (B / ROWS) * ROWS + (int)blockIdx.x; r < B; r += (int)gridDim.x) {
        const size_t off = (size_t)r * F + (size_t)tid;
        const float xv = x[off];
        const float ev = eps[off];
        int cnt = 0;
#pragma unroll
        for (int j = 0; j < NE; ++j) cnt += (xv >= e[j]) ? 1 : 0;
        const int idx = cnt - 1;
        int jc = idx < 0 ? 0 : (idx > NB - 1 ? NB - 1 : idx);
        const float s = s_scale[tid * NE + jc];
        const float noise = ((unsigned)idx < (unsigned)NB) ? ev * s : 0.0f;
        out[off] = xv + noise;
    }
}

extern "C" void kernel_launch(void* const* d_in, const int* in_sizes, int n_in,
                              void* d_out, int out_size, void* d_ws, size_t ws_size,
                              hipStream_t stream) {
    const float* x      = (const float*)d_in[0];   // [B, F]
    const float* bins   = (const float*)d_in[1];   // [F, NE]
    const float* eps    = (const float*)d_in[2];   // [B, F]
    const float* params = (const float*)d_in[3];   // [F*NB]
    float* out          = (float*)d_out;

    const int B = in_sizes[0] / F;

    int groups = (B + ROWS - 1) / ROWS;
    int grid = groups < 2048 ? (groups > 0 ? groups : 1) : 2048;

    noise_learn_kernel<<<grid, F, 0, stream>>>(x, bins, eps, params, out, B);
}